// GodSlayer2_25271587569864
// MI455X (gfx1250) — compile-verified
//
#include <hip/hip_runtime.h>
#include <math.h>

// MI455X / gfx1250 kernel: pairwise-Gram + 4-layer circular cross-stencil,
// fully LDS-resident per (batch, channel) tile.
//
//   grid  = 256 blocks  : one per (b, c), b = blk>>2, c = blk&3
//   block = 1024 threads: 32 wave32 waves per WGP
//   LDS   = 256 x 257 f32 tile (257-float stride -> bank-conflict free
//           for both row and column stencil accesses) + 32-float scratch
//
// Phase A: g = n·n^T via V_WMMA_F32_16X16X4_F32 (K=3 zero-padded to 4).
//          A fragment is wave-invariant (ti = wave>>1) and hoisted; loads
//          are branchless (unconditional b64+b32 + v_cndmask K-half select).
// Phase B: 4 layers of y = w0*x + sum_d wd*(x[i,j±d] + x[i±d,j]) (circular),
//          CELU, register-staged in-place update (2 barriers / layer).
// Phase C: LDS-resident reduction -> 1 float per block; second kernel folds
//          channels and applies exp(-s / 65536).

#define LTILE  256
#define STRIDE 257
#define NCH    4
#define NLAYER 4

typedef __attribute__((ext_vector_type(2))) float v2f;
typedef __attribute__((ext_vector_type(8))) float v8f;

__global__ __launch_bounds__(1024, 1)
void gram_stencil_kernel(const float* __restrict__ n,
                         const float* __restrict__ weights,
                         float* __restrict__ partial)
{
    __shared__ float xs[LTILE * STRIDE];   // 263,168 B
    __shared__ float red[32];

    const int bc = blockIdx.x;
    const int b  = bc >> 2;
    const int c  = bc & 3;

    const int tid  = threadIdx.x;
    const int wave = tid >> 5;
    const int lane = tid & 31;
    const bool hi  = (lane & 16) != 0;   // false -> K=0,1 ; true -> K=2,3
    const int l16  = lane & 15;

    // ---------------- Phase A: Gram matrix via WMMA f32 16x16x4 -------------
    // 256 output tiles of 16x16; 32 waves x 8 tiles each. Tiles 8w..8w+7
    // share one row-tile, so the A fragment is loop-invariant per wave.
    const float* nb  = n + (size_t)b * (LTILE * 3);
    const int   ti   = wave >> 1;          // row-tile (wave-uniform)
    const int   tj0  = (wave & 1) * 8;     // first col-tile of this wave

    const float* ra = nb + (ti * 16 + l16) * 3;
    const float a0 = ra[0], a1 = ra[1], a2 = ra[2];   // unconditional loads
    v2f A;
    A[0] = hi ? a2   : a0;                 // v_cndmask, no exec branches
    A[1] = hi ? 0.0f : a1;                 // K=3 zero pad

    const int  mrow    = (hi ? 8 : 0);     // M offset of this lane's C rows
    const bool diagTile_base = true;       // (ti==tj) checked per tile below

    #pragma unroll
    for (int t8 = 0; t8 < 8; ++t8) {
        const int tj = tj0 + t8;
        const float* rb = nb + (tj * 16 + l16) * 3;
        const float b0 = rb[0], b1 = rb[1], b2 = rb[2];
        v2f Bv;
        Bv[0] = hi ? b2   : b0;
        Bv[1] = hi ? 0.0f : b1;

        v8f Cz = {0.f, 0.f, 0.f, 0.f, 0.f, 0.f, 0.f, 0.f};
        v8f D = __builtin_amdgcn_wmma_f32_16x16x4_f32(
                    false, A, false, Bv, (short)0, Cz, false, false);

        // C/D layout: vgpr v, lanes 0-15 -> M=v, lanes 16-31 -> M=v+8; N=l16
        const bool diagTile = diagTile_base && (ti == tj);  // wave-uniform
        float* dst = xs + (ti * 16 + mrow) * STRIDE + tj * 16 + l16;
        #pragma unroll
        for (int v = 0; v < 8; ++v) {
            const bool onDiag = diagTile & ((v + mrow) == l16);
            dst[v * STRIDE] = onDiag ? 0.0f : D[v];
        }
    }
    __syncthreads();

    // ---------------- Phase B: 4 stencil layers, LDS-resident ---------------
    // Thread -> (row i, 64-column quadrant j0). A wave spans 32 consecutive
    // rows at one column -> stride 257 makes column loads conflict-free.
    const int i  = tid & 255;
    const int j0 = (tid >> 8) * 64;

    const float* rowp = xs + i * STRIDE;
    int rneg[4], rpos[4];
    #pragma unroll
    for (int d = 1; d <= 4; ++d) {
        rneg[d - 1] = ((i - d) & 255) * STRIDE;
        rpos[d - 1] = ((i + d) & 255) * STRIDE;
    }

    float y[64];
    for (int layer = 0; layer < NLAYER; ++layer) {
        const float* wp = weights + (size_t)(layer * NCH + c) * 5;  // uniform
        const float w0 = wp[0];
        const float wn[4] = { wp[1], wp[2], wp[3], wp[4] };

        #pragma unroll
        for (int e = 0; e < 64; ++e) {
            const int j = j0 + e;
            float v = w0 * rowp[j];
            #pragma unroll
            for (int d = 1; d <= 4; ++d) {
                const float h = rowp[(j - d) & 255] + rowp[(j + d) & 255]
                              + xs[rneg[d - 1] + j] + xs[rpos[d - 1] + j];
                v += wn[d - 1] * h;
            }
            y[e] = v;
        }
        __syncthreads();               // all reads of x done
        #pragma unroll
        for (int e = 0; e < 64; ++e) {
            const float v = y[e];
            xs[i * STRIDE + j0 + e] = (v > 0.0f) ? v : (__expf(v) - 1.0f);
        }
        __syncthreads();               // all writes visible for next layer
    }

    // ---------------- Phase C: reduce tile -> one partial per (b,c) ---------
    float acc = 0.0f;
    #pragma unroll
    for (int e = 0; e < 64; ++e)       // re-reads own writes; no barrier need
        acc += xs[i * STRIDE + j0 + e];

    #pragma unroll
    for (int off = 16; off > 0; off >>= 1)
        acc += __shfl_down(acc, off, 32);
    if (lane == 0) red[wave] = acc;
    __syncthreads();
    if (wave == 0) {
        float v = red[lane];           // exactly 32 waves -> 32 lanes
        #pragma unroll
        for (int off = 16; off > 0; off >>= 1)
            v += __shfl_down(v, off, 32);
        if (lane == 0) partial[bc] = v;
    }
}

__global__ void finalize_kernel(const float* __restrict__ partial,
                                float* __restrict__ out)
{
    const int t = threadIdx.x;
    if (t < 64) {
        const float* p = partial + t * 4;
        const float s = (p[0] + p[1]) + (p[2] + p[3]);
        out[t] = __expf(-s * (1.0f / 65536.0f));
    }
}

extern "C" void kernel_launch(void* const* d_in, const int* in_sizes, int n_in,
                              void* d_out, int out_size, void* d_ws, size_t ws_size,
                              hipStream_t stream)
{
    const float* n       = (const float*)d_in[0];   // (64, 256, 3) f32
    const float* weights = (const float*)d_in[1];   // (4, 4, 5)    f32
    float* partial = (float*)d_ws;                  // 256 floats of scratch
    float* out     = (float*)d_out;                 // 64 floats

    gram_stencil_kernel<<<256, 1024, 0, stream>>>(n, weights, partial);
    finalize_kernel<<<1, 64, 0, stream>>>(partial, out);
}